// YoloLayer_2095944040898
// MI455X (gfx1250) — compile-verified
//
#include <hip/hip_runtime.h>
#include <math.h>

// ---------------------------------------------------------------------------
// YOLO layer post-processing for MI455X (gfx1250).
// Memory-bound (188 MB @ 23.3 TB/s ~= 8 us floor). No GEMM -> no WMMA; the
// CDNA5 path exercised here is the Tensor Data Mover: each workgroup DMAs an
// 85-channel x 128-column strided tile (stride 5776 floats) into LDS with one
// tensor_load_to_lds (verified in asm: 2-SGPR-group 2-D form), waits on
// TENSORcnt, then does the elementwise math with an LDS softmax.
// VALU trimmed: hardware v_exp_f32 (__expf) + exp values cached in LDS so
// each class channel is exponentiated exactly once.
// ---------------------------------------------------------------------------

#define NUM_BA 48          // B*A = 16*3
#define HW_    5776        // 76*76
#define W_     76
#define TILE_W 128         // columns per workgroup (= blockDim.x, 4 waves)
#define NCH    85          // 5 + 80
#define NCLS   80
#define NTOT   (NUM_BA * HW_)   // 277248 columns

typedef unsigned int  v4u __attribute__((ext_vector_type(4)));
typedef int           v4i __attribute__((ext_vector_type(4)));
typedef int           v8i __attribute__((ext_vector_type(8)));

#if defined(__has_builtin)
#  if __has_builtin(__builtin_amdgcn_tensor_load_to_lds)
#    define HAVE_TDM 1
#  endif
#endif
#ifndef HAVE_TDM
#  define HAVE_TDM 0
#endif

// Hardware sigmoid: v_exp_f32 + rcp path; ~2 ULP, fine at softmax/threshold
// granularity.
__device__ __forceinline__ float sigm(float v) {
  return 1.0f / (1.0f + __expf(-v));
}

__global__ __launch_bounds__(TILE_W)
void yolo_layer_kernel(const float* __restrict__ xin, float* __restrict__ out) {
  // 85 channels * 128 columns * 4B = 43,520 B of LDS. Channel stride 128
  // floats -> consecutive lanes hit consecutive banks (conflict-free).
  __shared__ float tile[NCH * TILE_W];

  const int tid = threadIdx.x;
  const int tix = blockIdx.x;          // tile index within one (b,a) block
  const int ba  = blockIdx.y;          // 0..47  (= b*3 + a)
  const int x0  = tix * TILE_W;        // first column (hw) of this tile

#if HAVE_TDM
  if (tid < 32) {  // one wave issues the TDM op (EXEC is ignored by TDM)
    // Byte address of tile start: x + 4*((85*ba)*5776 + x0)
    unsigned long long gaddr =
        (unsigned long long)(const void*)xin +
        4ull * ((unsigned long long)ba * (unsigned long long)(NCH * HW_) +
                (unsigned long long)x0);
    // Low 32 bits of a generic LDS pointer == LDS byte offset (aperture rule).
    unsigned lds_off = (unsigned)(unsigned long long)(void*)&tile[0];
    unsigned rem = (unsigned)(HW_ - x0);  // valid width; TDM OOB reads -> 0

    // ---- D# group 0 (128b): count=1 | lds_addr | global_addr | type=2 ----
    v4u g0;
    g0[0] = 1u;                                   // count=1, user mode
    g0[1] = lds_off;                              // lds_addr (bytes)
    g0[2] = (unsigned)(gaddr & 0xFFFFFFFFull);    // global_addr[31:0]
    g0[3] = (unsigned)((gaddr >> 32) & 0x01FFFFFFull) | (2u << 30); // type=2

    // ---- D# group 1 (256b): 2-D tile, 4-byte elements ----
    v8i g1;
    g1[0] = (int)(2u << 16);                      // data_size=2 (4B); mask=0
    g1[1] = (int)((rem & 0xFFFFu) << 16);         // tensor_dim0[15:0]
    g1[2] = (int)((rem >> 16) |                   // tensor_dim0[31:16]
                  ((unsigned)NCH << 16));         // tensor_dim1[15:0] = 85
    g1[3] = (int)((unsigned)TILE_W << 16);        // dim1 hi=0 | tile_dim0=128
    g1[4] = (int)NCH;                             // tile_dim1=85, tile_dim2=0
    g1[5] = (int)HW_;                             // tensor_dim0_stride = 5776
    g1[6] = 0;                                    // stride0 hi | stride1 lo
    g1[7] = 0;                                    // stride1 hi

    v4i z4; z4[0] = z4[1] = z4[2] = z4[3] = 0;    // groups 2/3 unused (2-D)
#if __clang_major__ >= 23
    v8i z8; for (int i = 0; i < 8; ++i) z8[i] = 0;
    __builtin_amdgcn_tensor_load_to_lds(g0, g1, z4, z4, z8, 0);
#else
    __builtin_amdgcn_tensor_load_to_lds(g0, g1, z4, z4, 0);
#endif
    __builtin_amdgcn_s_wait_tensorcnt(0);         // DMA complete for this wave
  }
#else
  // Fallback: cooperative coalesced staging (still fully coalesced per row).
  for (int c = 0; c < NCH; ++c) {
    int col = x0 + tid;
    tile[c * TILE_W + tid] =
        (col < HW_) ? xin[((size_t)ba * NCH + c) * HW_ + col] : 0.0f;
  }
#endif
  __syncthreads();   // publish LDS tile to all 4 waves

  const int col = x0 + tid;
  if (col >= HW_) return;

  // Reference quirk: anchor vector is repeat(anchors, B*H*W) over the
  // (B,A,H,W) column order, so anchor index = n / (B*H*W) = ba >> 4.
  const int ai = ba >> 4;
  float aw, ah;
  if      (ai == 0) { aw = 116.0f; ah =  90.0f; }
  else if (ai == 1) { aw = 156.0f; ah = 198.0f; }
  else              { aw = 373.0f; ah = 326.0f; }

  const float v0 = tile[0 * TILE_W + tid];
  const float v1 = tile[1 * TILE_W + tid];
  const float v2 = tile[2 * TILE_W + tid];
  const float v3 = tile[3 * TILE_W + tid];
  const float v4 = tile[4 * TILE_W + tid];

  const float cx = (float)(col % W_);
  const float cy = (float)(col / W_);

  const float xv  = (sigm(v0) + cx) / 76.0f;
  const float yv  = (sigm(v1) + cy) / 76.0f;
  const float wv  = __expf(v2) * aw / 608.0f;
  const float hv  = __expf(v3) * ah / 608.0f;
  const float obj = sigm(v4);

  // Softmax over the 80 class channels, operating on this thread's private
  // LDS column (no cross-thread sharing after the barrier, so in-place
  // overwrite with exp values is race-free).
  // Pass 1: max.
  float m = -INFINITY;
#pragma unroll 8
  for (int c = 0; c < NCLS; ++c)
    m = fmaxf(m, tile[(5 + c) * TILE_W + tid]);

  // Pass 2: sum of exp; cache exp back into LDS so pass 3 needs no re-exp.
  float s = 0.0f;
#pragma unroll 8
  for (int c = 0; c < NCLS; ++c) {
    float e = __expf(tile[(5 + c) * TILE_W + tid] - m);
    tile[(5 + c) * TILE_W + tid] = e;
    s += e;
  }
  const float inv_s = 1.0f / s;

  const bool valid = (obj > 0.25f) && (xv <= 1.0f) && (yv <= 1.0f) &&
                     (wv <= 1.0f) && (hv <= 1.0f);
  const float vm = valid ? 1.0f : 0.0f;

  const long long n = (long long)ba * HW_ + col;
  const long long N = (long long)NTOT;

  // Coalesced stores: consecutive lanes -> consecutive n.
  out[0 * N + n] = xv  * vm;
  out[1 * N + n] = yv  * vm;
  out[2 * N + n] = wv  * vm;
  out[3 * N + n] = hv  * vm;
  out[4 * N + n] = obj * vm;

  const float k = obj * inv_s;
#pragma unroll 8
  for (int c = 0; c < NCLS; ++c) {
    float p = tile[(5 + c) * TILE_W + tid] * k;   // obj * softmax
    p = (p > 0.25f) ? p : 0.0f;
    out[(long long)(5 + c) * N + n] = p * vm;
  }
}

extern "C" void kernel_launch(void* const* d_in, const int* in_sizes, int n_in,
                              void* d_out, int out_size, void* d_ws, size_t ws_size,
                              hipStream_t stream) {
  (void)in_sizes; (void)n_in; (void)out_size; (void)d_ws; (void)ws_size;
  const float* x = (const float*)d_in[0];
  float* out = (float*)d_out;

  dim3 grid((HW_ + TILE_W - 1) / TILE_W, NUM_BA);  // (46, 48)
  yolo_layer_kernel<<<grid, TILE_W, 0, stream>>>(x, out);
}